// GraphEncoder_56925496541773
// MI455X (gfx1250) — compile-verified
//
#include <hip/hip_runtime.h>

typedef float v2f __attribute__((ext_vector_type(2)));
typedef float v8f __attribute__((ext_vector_type(8)));

// ---------------- degree / normalization ----------------

__global__ void fill_k(float* __restrict__ p, float v, int n) {
  int t = blockIdx.x * blockDim.x + threadIdx.x;
  if (t < n) p[t] = v;
}

__global__ void count_deg_k(const int* __restrict__ dst, float* __restrict__ deg, int E) {
  int t = blockIdx.x * blockDim.x + threadIdx.x;
  if (t < E) atomicAdd(&deg[dst[t]], 1.0f);
}

__global__ void rsqrt_k(float* __restrict__ p, int n) {
  int t = blockIdx.x * blockDim.x + threadIdx.x;
  if (t < n) p[t] = rsqrtf(p[t]);   // deg >= 1 always (self-loop)
}

// ---------------- fp32 WMMA GEMM with dis-scaling epilogue ----------------
// Y[i,:] = dis[i] * (X[i,:] @ W); value is dual-stored into AGG as well,
// which pre-seeds the self-loop contribution for the edge-aggregation pass.
// One wave computes one 16-row tile x CT adjacent 16-col tiles (A reuse).

template <int K, int D, int CT>
__global__ __launch_bounds__(256) void gemm_wmma_scaled(
    const float* __restrict__ X, const float* __restrict__ W,
    const float* __restrict__ dis, float* __restrict__ Y,
    float* __restrict__ AGG, int nRows) {
  const int lane      = threadIdx.x & 31;
  const int wave      = (int)((blockIdx.x * blockDim.x + threadIdx.x) >> 5);
  const int colGroups = D / (16 * CT);
  const int rowTiles  = (nRows + 15) >> 4;
  const int rowTile   = wave / colGroups;
  const int colGroup  = wave - rowTile * colGroups;
  if (rowTile >= rowTiles) return;        // wave-uniform: EXEC stays all-1s

  const int row0 = rowTile << 4;
  const int col0 = colGroup * (16 * CT);
  const int m    = lane & 15;             // A: row-in-tile; B/C: col-in-tile
  const int hi   = lane >> 4;             // selects K pair / output row half

  int ar = row0 + m;                      // clamp (not predicate) -> branch-free loads
  if (ar > nRows - 1) ar = nRows - 1;
  const float* aPtr = X + (size_t)ar * K + 2 * hi;
  const float* bCol = W + col0 + m;

  v8f acc[CT];
#pragma unroll
  for (int c = 0; c < CT; ++c) acc[c] = (v8f){};

#pragma unroll
  for (int k0 = 0; k0 < K; k0 += 4) {
    const v2f a = *(const v2f*)(aPtr + k0);            // one b64 load
    const float* bk = bCol + (size_t)(k0 + 2 * hi) * D;
#pragma unroll
    for (int c = 0; c < CT; ++c) {
      v2f b;
      b.x = bk[c * 16];
      b.y = bk[D + c * 16];
      acc[c] = __builtin_amdgcn_wmma_f32_16x16x4_f32(
          /*neg_a=*/false, a, /*neg_b=*/false, b,
          /*c_mod=*/(short)0, acc[c], /*reuse_a=*/false, /*reuse_b=*/false);
    }
  }

  // C/D 16x16 f32 layout: VGPR v -> row (v + 8*hi), col m
#pragma unroll
  for (int v = 0; v < 8; ++v) {
    const int r = row0 + v + 8 * hi;
    if (r < nRows) {
      const float dr = dis[r];
#pragma unroll
      for (int c = 0; c < CT; ++c) {
        const float val = dr * acc[c][v];
        const size_t off = (size_t)r * D + col0 + c * 16 + m;
        Y[off] = val;
        AGG[off] = val;
      }
    }
  }
}

// ---------------- edge aggregation: AGG[dst,:] += Y[src,:] ----------------
// Pure gather -> atomic scatter (normalization folded into GEMM/finalize).
// D/4 consecutive threads cover one edge row -> coalesced float4 gathers.

template <int D>
__global__ void edge_agg_k(const float* __restrict__ Y, const int* __restrict__ src,
                           const int* __restrict__ dst, float* __restrict__ AGG, int E) {
  constexpr int CH = D / 4;
  int t = blockIdx.x * blockDim.x + threadIdx.x;
  if (t >= E * CH) return;
  int e = t / CH;                      // compile-time divisor -> shift
  int c = t - e * CH;
  int s = src[e];
  int d = dst[e];
  float4 v = ((const float4*)(Y + (size_t)s * D))[c];
  float* o = AGG + (size_t)d * D + (size_t)c * 4;
  atomicAdd(o + 0, v.x);
  atomicAdd(o + 1, v.y);
  atomicAdd(o + 2, v.z);
  atomicAdd(o + 3, v.w);
}

// ---------------- finalize: OUT[i,j] = maybe_relu(dis[i]*AGG[i,j] + b[j]) --

template <int D, bool RELU>
__global__ void finalize_k(const float* __restrict__ AGG, const float* __restrict__ dis,
                           const float* __restrict__ bias, float* __restrict__ OUT, int N) {
  int t = blockIdx.x * blockDim.x + threadIdx.x;
  if (t >= N * D) return;
  int i = t / D;
  int j = t - i * D;
  float v = dis[i] * AGG[t] + bias[j];
  if (RELU) v = fmaxf(v, 0.0f);
  OUT[t] = v;
}

// ---------------- host orchestration ----------------

static inline int cdiv(long long a, long long b) { return (int)((a + b - 1) / b); }

extern "C" void kernel_launch(void* const* d_in, const int* in_sizes, int n_in,
                              void* d_out, int out_size, void* d_ws, size_t ws_size,
                              hipStream_t stream) {
  const float* x   = (const float*)d_in[0];
  const int*   ei  = (const int*)d_in[1];
  const float* W1  = (const float*)d_in[2];
  const float* b1  = (const float*)d_in[3];
  const float* W2  = (const float*)d_in[4];
  const float* b2  = (const float*)d_in[5];
  const float* Wmu = (const float*)d_in[6];
  const float* bmu = (const float*)d_in[7];
  const float* Wlv = (const float*)d_in[8];
  const float* blv = (const float*)d_in[9];

  const int IND = 128, HID = 64, LAT = 32;
  const int N = in_sizes[0] / IND;        // 50000
  const int E = in_sizes[1] / 2;          // 800000
  const int* src = ei;
  const int* dst = ei + E;

  float* ws   = (float*)d_ws;
  float* dis  = ws;                               // [N]
  float* bufA = ws + N;                           // [N*HID]
  float* bufB = bufA + (size_t)N * HID;           // [N*HID]
  float* bufC = bufB + (size_t)N * HID;           // [N*HID]
  float* mu   = (float*)d_out;                    // [N*LAT]
  float* lv   = mu + (size_t)N * LAT;             // [N*LAT]

  const int B = 256;
  const int rowTiles = (N + 15) >> 4;

  // --- normalization: dis = rsqrt(1 + indegree) ---
  fill_k<<<cdiv(N, B), B, 0, stream>>>(dis, 1.0f, N);
  count_deg_k<<<cdiv(E, B), B, 0, stream>>>(dst, dis, E);
  rsqrt_k<<<cdiv(N, B), B, 0, stream>>>(dis, N);

  // --- conv1: h1 = relu(dis .* agg(dis .* (x@W1)) + b1) ---
  gemm_wmma_scaled<128, 64, 2>
      <<<cdiv((long long)rowTiles * 2, 8), B, 0, stream>>>(x, W1, dis, bufA, bufB, N);
  edge_agg_k<64><<<cdiv((long long)E * 16, B), B, 0, stream>>>(bufA, src, dst, bufB, E);
  finalize_k<64, true>
      <<<cdiv((long long)N * HID, B), B, 0, stream>>>(bufB, dis, b1, bufB, N);  // h1 = bufB

  // --- conv2 ---
  gemm_wmma_scaled<64, 64, 2>
      <<<cdiv((long long)rowTiles * 2, 8), B, 0, stream>>>(bufB, W2, dis, bufC, bufA, N);
  edge_agg_k<64><<<cdiv((long long)E * 16, B), B, 0, stream>>>(bufC, src, dst, bufA, E);
  finalize_k<64, true>
      <<<cdiv((long long)N * HID, B), B, 0, stream>>>(bufA, dis, b2, bufA, N);  // h2 = bufA

  // --- heads: mu and logvar (Y tiles packed into bufC, AGG into bufB) ---
  float* yMu = bufC;
  float* yLv = bufC + (size_t)N * LAT;
  float* aMu = bufB;
  float* aLv = bufB + (size_t)N * LAT;
  gemm_wmma_scaled<64, 32, 2>
      <<<cdiv((long long)rowTiles, 8), B, 0, stream>>>(bufA, Wmu, dis, yMu, aMu, N);
  gemm_wmma_scaled<64, 32, 2>
      <<<cdiv((long long)rowTiles, 8), B, 0, stream>>>(bufA, Wlv, dis, yLv, aLv, N);

  edge_agg_k<32><<<cdiv((long long)E * 8, B), B, 0, stream>>>(yMu, src, dst, aMu, E);
  edge_agg_k<32><<<cdiv((long long)E * 8, B), B, 0, stream>>>(yLv, src, dst, aLv, E);

  finalize_k<32, false>
      <<<cdiv((long long)N * LAT, B), B, 0, stream>>>(aMu, dis, bmu, mu, N);
  finalize_k<32, false>
      <<<cdiv((long long)N * LAT, B), B, 0, stream>>>(aLv, dis, blv, lv, N);
}